// SO3OutputGrid_13417477832860
// MI455X (gfx1250) — compile-verified
//
#include <hip/hip_runtime.h>
#include <stdint.h>

typedef float v2f __attribute__((ext_vector_type(2)));
typedef float v8f __attribute__((ext_vector_type(8)));

#define BROT   1024
#define NGRID  294912
#define MTILES 64                    // 1024 / 16
#define NTILES 18432                 // 294912 / 16
#define NSPLIT 32
#define WAVES_PER_BLOCK 8
#define TILES_PER_WAVE (NTILES / (NSPLIT * WAVES_PER_BLOCK))   // 72

// Order-preserving float -> uint32 key (monotone for all finite floats).
__device__ __forceinline__ uint32_t fkey(float f) {
  uint32_t b = __float_as_uint(f);
  return (b & 0x80000000u) ? ~b : (b | 0x80000000u);
}

__global__ void so3_init_ws(unsigned long long* __restrict__ ws) {
  int i = blockIdx.x * blockDim.x + threadIdx.x;
  if (i < BROT) ws[i] = 0ull;   // key 0 == most-negative float
}

// Branch-free B-fragment gather for one 16-wide N tile.
// gp = grid + n*9 for this lane's column n; khalf = 0 (lanes 0-15) or 2.
// Chunk c=0: K = khalf,khalf+1 ; c=1: K = 4+khalf,5+khalf ; c=2: K=8 (x of
// low half only, everything else is zero padding of K 9..11).
__device__ __forceinline__ void load_bfrag(const float* __restrict__ gp,
                                           int khalf,
                                           v2f& b0, v2f& b1, float& b2x) {
  const float* p = gp + khalf;
  b0.x = p[0]; b0.y = p[1];
  b1.x = p[4]; b1.y = p[5];
  b2x  = gp[8];                 // in-bounds for ALL lanes; masked at use site
}

__global__ __launch_bounds__(256) void so3_sim_argmax(
    const float* __restrict__ rot,      // [1024, 9]
    const float* __restrict__ grid,     // [294912, 9]
    unsigned long long* __restrict__ ws // [1024] packed (key<<32 | ~idx)
) {
  const int lane  = threadIdx.x & 31;
  const int wave  = threadIdx.x >> 5;
  const int mbase = blockIdx.x * 16;
  const int m     = lane & 15;                 // row within A tile / col within D tile
  const int khalf = (lane >> 4) << 1;          // lanes 0-15: K+0,1 ; lanes 16-31: K+2,3

  // ---- A fragment: 16x12 (K padded 9 -> 12), three K=4 chunks -------------
  v2f a[3];
#pragma unroll
  for (int c = 0; c < 3; ++c) {
    const int k0 = 4 * c + khalf;
    a[c].x = (k0     < 9) ? rot[(mbase + m) * 9 + k0    ] : 0.f;
    a[c].y = (k0 + 1 < 9) ? rot[(mbase + m) * 9 + k0 + 1] : 0.f;
  }

  float best[8];
  int   bidx[8];
#pragma unroll
  for (int r = 0; r < 8; ++r) { best[r] = -3.0e38f; bidx[r] = 0; }

  const int tile0 = (blockIdx.y * WAVES_PER_BLOCK + wave) * TILES_PER_WAVE;
  const int n0    = tile0 * 16 + m;

  // ---- software pipeline: B fragment for tile t+1 loads during tile t -----
  v2f   nb0, nb1;
  float nb2;
  load_bfrag(grid + n0 * 9, khalf, nb0, nb1, nb2);

  for (int t = 0; t < TILES_PER_WAVE; ++t) {
    const int n = n0 + t * 16;                 // this lane's candidate column
    const v2f   b0 = nb0;
    const v2f   b1 = nb1;
    const float b2 = nb2;

    if (t + 1 < TILES_PER_WAVE) {              // uniform condition: no exec divergence
      const float* gnext = grid + (n + 16) * 9;
      __builtin_prefetch(gnext + 16 * 9, 0, 1);   // tile t+2 -> global_prefetch_b8
      load_bfrag(gnext, khalf, nb0, nb1, nb2);
    }

    v2f b2f;                                   // chunk c=2, zero-padded
    b2f.x = khalf ? 0.f : b2;                  // single v_cndmask, no branch
    b2f.y = 0.f;

    // ---- 16x16 sim tile via 3 chained f32 WMMAs ---------------------------
    v8f d = {0.f, 0.f, 0.f, 0.f, 0.f, 0.f, 0.f, 0.f};
    d = __builtin_amdgcn_wmma_f32_16x16x4_f32(false, a[0], false, b0,
                                              (short)0, d, false, false);
    d = __builtin_amdgcn_wmma_f32_16x16x4_f32(false, a[1], false, b1,
                                              (short)0, d, false, false);
    d = __builtin_amdgcn_wmma_f32_16x16x4_f32(false, a[2], false, b2f,
                                              (short)0, d, false, false);

    // ---- running argmax: slot r holds row M=r (lanes 0-15) / M=r+8 --------
#pragma unroll
    for (int r = 0; r < 8; ++r) {
      const float v = d[r];
      if (v > best[r]) { best[r] = v; bidx[r] = n; }
    }
  }

  // ---- butterfly argmax across the 16 lanes sharing each row --------------
#pragma unroll
  for (int r = 0; r < 8; ++r) {
    float v = best[r];
    int   i = bidx[r];
#pragma unroll
    for (int off = 8; off >= 1; off >>= 1) {
      const float ov = __shfl_xor(v, off, 32);
      const int   oi = __shfl_xor(i, off, 32);
      if (ov > v || (ov == v && oi < i)) { v = ov; i = oi; }
    }
    best[r] = v; bidx[r] = i;
  }

  // lanes 0 / 16 publish rows mbase+0..7 / mbase+8..15
  if (lane == 0 || lane == 16) {
    const int boff = mbase + ((lane >> 4) << 3);
#pragma unroll
    for (int r = 0; r < 8; ++r) {
      const unsigned long long packed =
          ((unsigned long long)fkey(best[r]) << 32) |
          (unsigned long long)(~(uint32_t)bidx[r]);   // ~idx: ties pick lowest idx
      atomicMax(&ws[boff + r], packed);
    }
  }
}

__global__ void so3_finalize(const unsigned long long* __restrict__ ws,
                             const float* __restrict__ grid,
                             float* __restrict__ out) {
  const int b = blockIdx.x * blockDim.x + threadIdx.x;
  if (b >= BROT) return;
  const unsigned long long p = ws[b];
  const uint32_t key  = (uint32_t)(p >> 32);
  const uint32_t idx  = ~(uint32_t)p;
  const uint32_t bits = (key & 0x80000000u) ? (key & 0x7FFFFFFFu) : ~key;
  out[b] = __uint_as_float(bits);                      // dot_trace
  const float* g  = grid + idx * 9;
  float*       nb = out + BROT + b * 9;                // nearest
#pragma unroll
  for (int j = 0; j < 9; ++j) nb[j] = g[j];
}

extern "C" void kernel_launch(void* const* d_in, const int* in_sizes, int n_in,
                              void* d_out, int out_size, void* d_ws, size_t ws_size,
                              hipStream_t stream) {
  const float* rot  = (const float*)d_in[0];   // [1024*9]
  const float* grid = (const float*)d_in[1];   // [294912*9]
  float* out = (float*)d_out;                  // [1024 + 1024*9]
  unsigned long long* ws = (unsigned long long*)d_ws;

  so3_init_ws<<<4, 256, 0, stream>>>(ws);
  dim3 g(MTILES, NSPLIT);
  so3_sim_argmax<<<g, 256, 0, stream>>>(rot, grid, ws);
  so3_finalize<<<4, 256, 0, stream>>>(ws, grid, out);
}